// GraphConvLayer_48550310314068
// MI455X (gfx1250) — compile-verified
//
#include <hip/hip_runtime.h>

// GraphConv forward for MI455X (gfx1250, wave32):
//   1) support = feature @ W        -> V_WMMA_F32_16X16X4_F32 (fp32 matrix pipe)
//   2) out[dst] += w_e * support[src] -> per-edge wave, float4 + global_atomic_add_f32
//   3) out = dropout(relu(out + bias))  (deterministic hash mask, inverted scaling)

typedef float v2f __attribute__((ext_vector_type(2)));
typedef float v8f __attribute__((ext_vector_type(8)));

#define D 128            // D_IN == D_OUT == 128
#define LDS_STRIDE 132   // 128 + 4 pad -> bank-conflict-free column reads (132 % 64 == 4)

// ---------------------------------------------------------------------------
// 1) GEMM: one 16-row M-tile per block; 8 waves each own a 16-col N-tile.
//    K=128 -> 32 chained V_WMMA_F32_16X16X4_F32 per wave.
// ---------------------------------------------------------------------------
__global__ __launch_bounds__(256) void gcn_gemm_wmma(const float* __restrict__ A,
                                                     const float* __restrict__ W,
                                                     float* __restrict__ S) {
    __shared__ float ldsA[16 * LDS_STRIDE];
    const int m0  = blockIdx.x * 16;
    const int tid = threadIdx.x;

    // Stage the 16x128 A tile into LDS, fully coalesced (each thread: 2x float4).
    {
        const int r = tid >> 4;            // 0..15 rows
        const int c = (tid & 15) * 8;      // 8 floats per thread
        const float4* g = (const float4*)(A + (size_t)(m0 + r) * D + c);
        float4 p0 = g[0];
        float4 p1 = g[1];
        float* l = ldsA + r * LDS_STRIDE + c;
        *(float4*)(l)     = p0;
        *(float4*)(l + 4) = p1;
    }
    __syncthreads();

    const int lane = tid & 31;
    const int wave = tid >> 5;
    const int n0   = wave * 16;     // 8 waves cover N = 0..127
    const int h    = lane >> 4;     // lane half (ISA: K offset 2h, C row offset 8h)
    const int mr   = lane & 15;     // M for A-frag / N for B-frag / col for C

    // W stays resident in WGP$/L2 (64 KB total); hint the prefetcher.
    __builtin_prefetch(W + n0 + mr, 0, 3);

    v8f acc = {};
    const float* abase = ldsA + mr * LDS_STRIDE + 2 * h;   // float2 => ds_load_b64
    const float* wbase = W + n0 + mr;

#pragma unroll
    for (int kk = 0; kk < D; kk += 4) {
        // A frag: VGPR v, half h -> K = kk + v + 2h  (contiguous pair in LDS)
        v2f a = *(const v2f*)(abase + kk);
        // B frag: VGPR v, half h -> row K = kk + v + 2h, col n0+mr
        v2f b;
        b.x = wbase[(size_t)(kk + 2 * h)     * D];
        b.y = wbase[(size_t)(kk + 2 * h + 1) * D];
        acc = __builtin_amdgcn_wmma_f32_16x16x4_f32(false, a, false, b,
                                                    (short)0, acc, false, false);
    }

    // C/D layout: VGPR r -> row (r + 8h), col = mr
    float* out = S + (size_t)m0 * D + n0 + mr;
#pragma unroll
    for (int r = 0; r < 8; ++r)
        out[(size_t)(r + 8 * h) * D] = acc[r];
}

// Scalar tail for M not divisible by 16 (empty for N_NODES = 100000).
__global__ void gcn_gemm_tail(const float* __restrict__ A, const float* __restrict__ W,
                              float* __restrict__ S, int start, int M) {
    const int row = start + blockIdx.x;
    if (row >= M) return;
    const int n = threadIdx.x;  // 128 threads = one output column each
    float acc = 0.f;
    for (int k = 0; k < D; ++k)
        acc = fmaf(A[(size_t)row * D + k], W[(size_t)k * D + n], acc);
    S[(size_t)row * D + n] = acc;
}

// ---------------------------------------------------------------------------
// 2) zero-init + edge scatter (segment-sum via hardware f32 atomics in L2)
// ---------------------------------------------------------------------------
__global__ void gcn_zero4(float4* __restrict__ p, int n4) {
    const int i = blockIdx.x * blockDim.x + threadIdx.x;
    if (i < n4) p[i] = make_float4(0.f, 0.f, 0.f, 0.f);
}

__global__ __launch_bounds__(256) void gcn_spmm(const float* __restrict__ support,
                                                const float* __restrict__ ew,
                                                const int* __restrict__ esrc,
                                                const int* __restrict__ edst,
                                                float* __restrict__ out, int E) {
    const int e = blockIdx.x * 8 + (int)(threadIdx.x >> 5);  // 8 edges / 256-thr block
    if (e >= E) return;                                       // wave-uniform exit
    const int lane = threadIdx.x & 31;
    const int s = esrc[e];
    const int d = edst[e];
    const float w = ew[e];
    // 32 lanes x float4 = the full 128-float row (global_load_b128)
    const float4 v = ((const float4*)(support + (size_t)s * D))[lane];
    float* o = out + (size_t)d * D + lane * 4;
    unsafeAtomicAdd(o + 0, v.x * w);   // global_atomic_add_f32, no-return
    unsafeAtomicAdd(o + 1, v.y * w);
    unsafeAtomicAdd(o + 2, v.z * w);
    unsafeAtomicAdd(o + 3, v.w * w);
}

// ---------------------------------------------------------------------------
// 3) epilogue: bias + ReLU + deterministic dropout (p=0.2, inverted scaling)
// ---------------------------------------------------------------------------
__device__ __forceinline__ unsigned gcn_hash(unsigned x) {
    x ^= 0x9E3779B9u;
    x ^= x >> 16; x *= 0x7FEB352Du;
    x ^= x >> 15; x *= 0x846CA68Bu;
    x ^= x >> 16;
    return x;
}

__global__ void gcn_epilogue(float* __restrict__ out, const float* __restrict__ bias, int n4) {
    const int i = blockIdx.x * blockDim.x + threadIdx.x;
    if (i >= n4) return;
    const int base = i * 4;
    const int col  = base & (D - 1);          // 4-aligned, so 4 cols in same row
    float4 v = ((float4*)out)[i];
    const float4 b = *(const float4*)(bias + col);
    float r[4] = { v.x + b.x, v.y + b.y, v.z + b.z, v.w + b.w };
    const unsigned keep_thresh = 3435973836u; // 0.8 * 2^32
#pragma unroll
    for (int j = 0; j < 4; ++j) {
        float x = r[j] > 0.f ? r[j] : 0.f;
        r[j] = (gcn_hash((unsigned)(base + j)) < keep_thresh) ? x * 1.25f : 0.f;
    }
    ((float4*)out)[i] = make_float4(r[0], r[1], r[2], r[3]);
}

// ---------------------------------------------------------------------------
extern "C" void kernel_launch(void* const* d_in, const int* in_sizes, int n_in,
                              void* d_out, int out_size, void* d_ws, size_t ws_size,
                              hipStream_t stream) {
    const float* feature = (const float*)d_in[0];
    const float* eweight = (const float*)d_in[1];
    const float* weight  = (const float*)d_in[2];
    const float* bias    = (const float*)d_in[3];
    const int*   esrc    = (const int*)d_in[4];
    const int*   edst    = (const int*)d_in[5];

    const int N = in_sizes[0] / D;   // 100000 nodes
    const int E = in_sizes[1];       // 1.6M edges

    float* out     = (float*)d_out;
    float* support = (float*)d_ws;   // N*D floats = 51.2 MB scratch

    // 1) support = feature @ W  (WMMA fp32)
    const int mt = N / 16;
    if (mt > 0)
        gcn_gemm_wmma<<<mt, 256, 0, stream>>>(feature, weight, support);
    const int tail = N - mt * 16;
    if (tail > 0)
        gcn_gemm_tail<<<tail, D, 0, stream>>>(feature, weight, support, mt * 16, N);

    // 2) out = segment_sum over edges
    const int n4 = (N * D) / 4;
    gcn_zero4<<<(n4 + 255) / 256, 256, 0, stream>>>((float4*)out, n4);
    gcn_spmm<<<(E + 7) / 8, 256, 0, stream>>>(support, eweight, esrc, edst, out, E);

    // 3) bias + ReLU + dropout
    gcn_epilogue<<<(n4 + 255) / 256, 256, 0, stream>>>(out, bias, n4);
}